// FreqChunker_14413910245440
// MI455X (gfx1250) — compile-verified
//
#include <hip/hip_runtime.h>

#define SEQ_LEN 4096
#define TILE 32
#define RANK_FIRST 1996
#define LOG_THRESHOLD (-20.0f)

// CDNA5 async global->LDS copy path (ASYNCcnt-tracked), with safe fallback.
#if defined(__gfx1250__) && __has_builtin(__builtin_amdgcn_global_load_async_to_lds_b32) && \
    __has_builtin(__builtin_amdgcn_s_wait_asynccnt)
#define USE_ASYNC_LDS 1
#else
#define USE_ASYNC_LDS 0
#endif

__device__ __forceinline__ float zipf_ll(int id) {
  int r = id - (RANK_FIRST - 1);
  float rank = (float)(r > 1 ? r : 1);
  return -logf(rank);  // alpha = 1.0
}

__global__ __launch_bounds__(TILE) void FreqChunker_seg_kernel(
    const int* __restrict__ ids, const int* __restrict__ msk,
    int* __restrict__ out) {
  const int row  = blockIdx.x;
  const int lane = threadIdx.x;  // wave32: one wave per row
  const long rb  = (long)row * SEQ_LEN;
  const int* idsR = ids + rb;
  const int* mskR = msk + rb;
  int* outR = out + rb;

  __shared__ int lds_ids[2][TILE];
  __shared__ int lds_msk[2][TILE];

#if USE_ASYNC_LDS
  // Prologue: stage tile 0 via the async copy engine (per-lane 4B -> 128B/wave).
  __builtin_amdgcn_global_load_async_to_lds_b32(
      (int*)(idsR + lane), &lds_ids[0][lane], 0, 0);
  __builtin_amdgcn_global_load_async_to_lds_b32(
      (int*)(mskR + lane), &lds_msk[0][lane], 0, 0);
#endif

  // Uniform (replicated across lanes) sequential-scan state.
  float acc     = 0.0f;  // sums[j] - sums[left]
  float ll_prev = 0.0f;  // ll[j-1]
  int   left_m  = 1;
  int   seg     = 0;

  const int ntiles = SEQ_LEN / TILE;
  for (int t = 0; t < ntiles; ++t) {
    const int base = t * TILE;
    const int buf  = t & 1;

    int id_lane, m_lane;
#if USE_ASYNC_LDS
    if (t + 1 < ntiles) {
      const int nb = buf ^ 1;
      __builtin_amdgcn_global_load_async_to_lds_b32(
          (int*)(idsR + base + TILE + lane), &lds_ids[nb][lane], 0, 0);
      __builtin_amdgcn_global_load_async_to_lds_b32(
          (int*)(mskR + base + TILE + lane), &lds_msk[nb][lane], 0, 0);
      __builtin_amdgcn_s_wait_asynccnt(2);  // current tile's 2 ops complete
    } else {
      __builtin_amdgcn_s_wait_asynccnt(0);
    }
    asm volatile("" ::: "memory");  // keep LDS reads below the async wait
    id_lane = lds_ids[buf][lane];
    m_lane  = lds_msk[buf][lane];
#else
    (void)buf;
    if (t + 1 < ntiles) {
      __builtin_prefetch(idsR + base + TILE + lane, 0, 0);
      __builtin_prefetch(mskR + base + TILE + lane, 0, 0);
    }
    id_lane = idsR[base + lane];
    m_lane  = mskR[base + lane];
#endif

    const float ll_lane = zipf_ll(id_lane);

    // Sequential 32-step scan over the tile; all lanes replicate the uniform
    // state, lane k captures its own segment id when j == base + k.
    int myseg = 0;
    for (int k = 0; k < TILE; ++k) {
      const float llk = __shfl(ll_lane, k, TILE);
      const int   mk  = __shfl(m_lane, k, TILE);
      const int   j   = base + k;
      if (j > 0) {
        acc += ll_prev;
        const bool boundary =
            (left_m == 0) || (acc < LOG_THRESHOLD) || (mk == 0);
        seg += boundary ? 1 : 0;
        if (boundary) { acc = 0.0f; left_m = mk; }
      } else {
        left_m = mk;  // j == 0: seg stays 0, acc stays 0
      }
      ll_prev = llk;
      myseg = (k == lane) ? seg : myseg;
    }
    outR[base + lane] = myseg;
  }
}

extern "C" void kernel_launch(void* const* d_in, const int* in_sizes, int n_in,
                              void* d_out, int out_size, void* d_ws, size_t ws_size,
                              hipStream_t stream) {
  (void)n_in; (void)out_size; (void)d_ws; (void)ws_size;
  const int* token_ids = (const int*)d_in[0];
  const int* reg_mask  = (const int*)d_in[1];
  int* out = (int*)d_out;  // reference output dtype is int32 (4-byte elems)
  const int total = in_sizes[0];
  const int B = total / SEQ_LEN;  // 32 rows of 4096
  FreqChunker_seg_kernel<<<dim3(B), dim3(TILE), 0, stream>>>(token_ids, reg_mask, out);
}